// QuantumLayer_21397527069313
// MI455X (gfx1250) — compile-verified
//
#include <hip/hip_runtime.h>
#include <hip/hip_bf16.h>

// MI455X / gfx1250, wave32. One wave simulates one 10-qubit statevector fully
// in registers (re[32], im[32] per lane; amp index = jj*32 + lane).
// Readout expectation values computed with V_WMMA_F32_16X16X4_F32.

#define DEVINL __device__ __forceinline__

typedef __attribute__((ext_vector_type(2))) float v2f;
typedef __attribute__((ext_vector_type(8))) float v8f;

struct C2 { float r00,i00,r01,i01,r10,i10,r11,i11; };

// Fused per-wire gate: G = RZ(w2) * RY(w1) * RZ(w0) * RY(theta)
// cy = cos(theta/2), sy = sin(theta/2)
DEVINL void build_gate(float cy, float sy, float w0, float w1, float w2, C2& g) {
  float c0 = __cosf(0.5f * w0), s0 = __sinf(0.5f * w0);
  float c1 = __cosf(0.5f * w1), s1 = __sinf(0.5f * w1);
  float c2 = __cosf(0.5f * w2), s2 = __sinf(0.5f * w2);
  // A = RZ(w0) * RY(theta): row0 *= e^{-i w0/2}, row1 *= e^{+i w0/2}
  float a00r =  cy * c0, a00i = -cy * s0;
  float a01r = -sy * c0, a01i =  sy * s0;
  float a10r =  sy * c0, a10i =  sy * s0;
  float a11r =  cy * c0, a11i =  cy * s0;
  // B = RY(w1) * A
  float b00r = c1*a00r - s1*a10r, b00i = c1*a00i - s1*a10i;
  float b01r = c1*a01r - s1*a11r, b01i = c1*a01i - s1*a11i;
  float b10r = s1*a00r + c1*a10r, b10i = s1*a00i + c1*a10i;
  float b11r = s1*a01r + c1*a11r, b11i = s1*a01i + c1*a11i;
  // G = RZ(w2) * B: row0 * (c2 - i s2), row1 * (c2 + i s2)
  g.r00 = b00r*c2 + b00i*s2;  g.i00 = b00i*c2 - b00r*s2;
  g.r01 = b01r*c2 + b01i*s2;  g.i01 = b01i*c2 - b01r*s2;
  g.r10 = b10r*c2 - b10i*s2;  g.i10 = b10i*c2 + b10r*s2;
  g.r11 = b11r*c2 - b11i*s2;  g.i11 = b11i*c2 + b11r*s2;
}

// Wires 0..4: pairs within lane, jj-stride T = 1<<(4-wire).
template<int T>
DEVINL void gate_inlane(float* re, float* im, const C2& g) {
#pragma unroll
  for (int j0 = 0; j0 < 32; j0 += 2 * T) {
#pragma unroll
    for (int j = 0; j < T; ++j) {
      const int p = j0 + j, q = p + T;
      float ar = re[p], ai = im[p], br = re[q], bi = im[q];
      re[p] = g.r00*ar - g.i00*ai + g.r01*br - g.i01*bi;
      im[p] = g.r00*ai + g.i00*ar + g.r01*bi + g.i01*br;
      re[q] = g.r10*ar - g.i10*ai + g.r11*br - g.i11*bi;
      im[q] = g.r10*ai + g.i10*ar + g.r11*bi + g.i11*br;
    }
  }
}

// Wires 5..9: partner amplitude is in lane ^ (1<<MB).
template<int MB>
DEVINL void gate_xlane(float* re, float* im, const C2& g, int lane) {
  const int bit = (lane >> MB) & 1;
  const float owr = bit ? g.r11 : g.r00;   // coeff on own amplitude
  const float owi = bit ? g.i11 : g.i00;
  const float par = bit ? g.r10 : g.r01;   // coeff on partner amplitude
  const float pai = bit ? g.i10 : g.i01;
#pragma unroll
  for (int j = 0; j < 32; ++j) {
    float br = __shfl_xor(re[j], 1 << MB, 32);
    float bi = __shfl_xor(im[j], 1 << MB, 32);
    float nr = owr*re[j] - owi*im[j] + par*br - pai*bi;
    float ni = owr*im[j] + owi*re[j] + par*bi + pai*br;
    re[j] = nr; im[j] = ni;
  }
}

template<int W>
DEVINL void apply_wire(float* re, float* im, const C2& g, int lane) {
  if constexpr (W < 5) gate_inlane<(1 << (4 - W))>(re, im, g);
  else                 gate_xlane<9 - W>(re, im, g, lane);
}

// CNOT with both control (jj bit CB) and target (jj bit TB) in-lane:
// pure compile-time register permutation (free after unroll).
template<int CB, int TB>
DEVINL void cnot_jj(float* re, float* im) {
#pragma unroll
  for (int j = 0; j < 32; ++j) {
    if (((j >> CB) & 1) && !((j >> TB) & 1)) {
      const int q = j | (1 << TB);
      float t = re[j]; re[j] = re[q]; re[q] = t;
      t = im[j]; im[j] = im[q]; im[q] = t;
    }
  }
}

// CNOT(4 -> 5): control = jj bit 0, target = lane bit 4. For odd jj the
// exchange across lane^16 is unconditional (both partners have ctrl=1).
DEVINL void cnot_4_5(float* re, float* im) {
#pragma unroll
  for (int j = 1; j < 32; j += 2) {
    re[j] = __shfl_xor(re[j], 16, 32);
    im[j] = __shfl_xor(im[j], 16, 32);
  }
}

// CNOT with control lane bit CB, target lane bit TB: predicated lane swap.
template<int CB, int TB>
DEVINL void cnot_lane(float* re, float* im, int lane) {
  const bool c = (lane >> CB) & 1;
#pragma unroll
  for (int j = 0; j < 32; ++j) {
    float pr = __shfl_xor(re[j], 1 << TB, 32);
    float pi = __shfl_xor(im[j], 1 << TB, 32);
    re[j] = c ? pr : re[j];
    im[j] = c ? pi : im[j];
  }
}

// CNOT(9 -> 0): control lane bit 0, target jj bit 4: predicated register swap.
DEVINL void cnot_9_0(float* re, float* im, int lane) {
  const bool c = lane & 1;
#pragma unroll
  for (int j = 0; j < 16; ++j) {
    float t0 = re[j], t1 = re[j + 16];
    re[j]      = c ? t1 : t0;
    re[j + 16] = c ? t0 : t1;
    t0 = im[j]; t1 = im[j + 16];
    im[j]      = c ? t1 : t0;
    im[j + 16] = c ? t0 : t1;
  }
}

__global__ __launch_bounds__(512)
void qsim_kernel(const float* __restrict__ x, const float* __restrict__ w,
                 float* __restrict__ out) {
  __shared__ float Alds[16 * 200];   // 16 batch rows x (K=192 + pad 8)
  __shared__ float tile[256];        // 16x16 f32 output accumulation tile

  const int lane = threadIdx.x & 31;
  const int wave = threadIdx.x >> 5;
  const int b    = blockIdx.x * 16 + wave;

  // Data-encoding half-angles: theta/2 = tanh(x)*pi/2 (computed once, reused
  // every layer).
  float cy[10], sy[10];
#pragma unroll
  for (int i = 0; i < 10; ++i) {
    const float th = tanhf(x[b * 10 + i]) * 1.57079632679489662f;
    cy[i] = __cosf(th);
    sy[i] = __sinf(th);
  }

  // |0...0>
  float re[32], im[32];
#pragma unroll
  for (int j = 0; j < 32; ++j) { re[j] = 0.0f; im[j] = 0.0f; }
  re[0] = (lane == 0) ? 1.0f : 0.0f;

  for (int l = 0; l < 4; ++l) {
    const float* wl = w + l * 30;
    C2 g;
#define DO_WIRE(I)                                                        \
    build_gate(cy[I], sy[I], wl[(I)*3+0], wl[(I)*3+1], wl[(I)*3+2], g);   \
    apply_wire<I>(re, im, g, lane);
    DO_WIRE(0) DO_WIRE(1) DO_WIRE(2) DO_WIRE(3) DO_WIRE(4)
    DO_WIRE(5) DO_WIRE(6) DO_WIRE(7) DO_WIRE(8) DO_WIRE(9)
#undef DO_WIRE
    // CNOT chain 0->1 ... 8->9, then 9->0 (order matters; kept sequential)
    cnot_jj<4, 3>(re, im);
    cnot_jj<3, 2>(re, im);
    cnot_jj<2, 1>(re, im);
    cnot_jj<1, 0>(re, im);
    cnot_4_5(re, im);
    cnot_lane<4, 3>(re, im, lane);
    cnot_lane<3, 2>(re, im, lane);
    cnot_lane<2, 1>(re, im, lane);
    cnot_lane<1, 0>(re, im, lane);
    cnot_9_0(re, im, lane);
  }

  // Per-lane partial reductions of probabilities:
  //   P   = sum_j p[j]                       (for lane-bit wires 5..9)
  //   Q_w = sum_j sign(jj bit 4-w) * p[j]    (for jj-bit wires 0..4)
  float P = 0.0f, Q0 = 0.0f, Q1 = 0.0f, Q2 = 0.0f, Q3 = 0.0f, Q4 = 0.0f;
#pragma unroll
  for (int j = 0; j < 32; ++j) {
    const float p = re[j] * re[j] + im[j] * im[j];
    P  += p;
    Q0 += ((j >> 4) & 1) ? -p : p;
    Q1 += ((j >> 3) & 1) ? -p : p;
    Q2 += ((j >> 2) & 1) ? -p : p;
    Q3 += ((j >> 1) & 1) ? -p : p;
    Q4 += ( j       & 1) ? -p : p;
  }
  float* Arow = Alds + wave * 200;
  Arow[       lane] = P;
  Arow[ 32 +  lane] = Q0;
  Arow[ 64 +  lane] = Q1;
  Arow[ 96 +  lane] = Q2;
  Arow[128 +  lane] = Q3;
  Arow[160 +  lane] = Q4;

  if (threadIdx.x < 256) tile[threadIdx.x] = 0.0f;
  __syncthreads();

  // Readout GEMM: D(16x16) = A(16x192) * B(192x16) via v_wmma_f32_16x16x4_f32.
  // 48 K-chunks of 4 split over 16 waves (3 WMMAs each). EXEC is all ones at
  // every WMMA (no divergent control flow in this section).
  // A frag: lanes 0-15 hold M=lane, K = k0+{0,1}; lanes 16-31 K = k0+{2,3}.
  //
  // B matrix (K=192, branch-free synthesis):
  //   k <  32 : columns 5..9 carry sign of bit (9-n) of k (lane-bit wires)
  //   k >= 32 : block w' = k/32 - 1 is a delta column selecting Q_{w'} sums
  const int mrow  = lane & 15;                 // A row / B,C column index n
  const int khalf = (lane >> 4) * 2;
  // Loop-invariant per-lane column predicates (selects, not branches):
  const bool actv  = ((unsigned)(mrow - 5)) < 5u;   // 5 <= n < 10
  const float cmsk = actv ? 1.0f : 0.0f;            // column live in block 0
  const int  shf   = actv ? (9 - mrow) : 0;         // bit position for sign
  const int  k0l   = wave * 12 + khalf;             // this lane's first K

  v8f acc = {0.f, 0.f, 0.f, 0.f, 0.f, 0.f, 0.f, 0.f};
#pragma unroll
  for (int t = 0; t < 3; ++t) {
    const int k   = k0l + 4 * t;
    const int blk = k >> 5;                     // chunk never crosses a block
    const float isb0 = (blk == 0) ? 1.0f : 0.0f;
    const float dlt  = (mrow == blk - 1) ? 1.0f : 0.0f;
    const float sx = 1.0f - 2.0f * (float)(((k     & 31) >> shf) & 1);
    const float sy2= 1.0f - 2.0f * (float)((((k+1) & 31) >> shf) & 1);
    v2f a, bb;
    a.x  = Alds[mrow * 200 + k];
    a.y  = Alds[mrow * 200 + k + 1];
    bb.x = isb0 * (cmsk * sx ) + (1.0f - isb0) * dlt;
    bb.y = isb0 * (cmsk * sy2) + (1.0f - isb0) * dlt;
    acc = __builtin_amdgcn_wmma_f32_16x16x4_f32(
        /*neg_a=*/false, a, /*neg_b=*/false, bb,
        /*c_mod=*/(short)0, acc, /*reuse_a=*/false, /*reuse_b=*/false);
  }
  // Reduce the 16 per-wave partial tiles with LDS float atomics (ds_add_f32).
  const int mbase = (lane >> 4) * 8;
#pragma unroll
  for (int r = 0; r < 8; ++r)
    atomicAdd(&tile[(mbase + r) * 16 + mrow], acc[r]);
  __syncthreads();

  if (threadIdx.x < 160) {
    const int m = threadIdx.x / 10, n = threadIdx.x % 10;
    out[(blockIdx.x * 16 + m) * 10 + n] = tile[m * 16 + n];
  }
}

extern "C" void kernel_launch(void* const* d_in, const int* in_sizes, int n_in,
                              void* d_out, int out_size, void* d_ws, size_t ws_size,
                              hipStream_t stream) {
  const float* x = (const float*)d_in[0];   // (16384, 10) f32
  const float* w = (const float*)d_in[1];   // (4, 10, 3) f32
  float* out = (float*)d_out;               // (16384, 10) f32
  const int B = in_sizes[0] / 10;           // 16384, multiple of 16
  qsim_kernel<<<dim3(B / 16), dim3(512), 0, stream>>>(x, w, out);
}